// Pooling_85624468013343
// MI455X (gfx1250) — compile-verified
//
#include <hip/hip_runtime.h>

// CDNA5 / gfx1250 pooling-with-learned-powers kernel.
// Memory-bound (512MB h read @ 23.3 TB/s ~ 22us floor). Score GEMV done with
// V_WMMA_F32_16X16X4_F32 (weights folded into A, B == 1 so the result is
// independent of the B fragment K-layout). Counts via binary search on the
// sorted seg_ids (no atomics); output segment-sum via run-aggregated atomics.

typedef float v2f __attribute__((ext_vector_type(2)));
typedef float v8f __attribute__((ext_vector_type(8)));

#if __has_builtin(__builtin_amdgcn_exp2f)
#define EXP2F(x) __builtin_amdgcn_exp2f(x)
#else
#define EXP2F(x) exp2f(x)
#endif

__global__ void zero_out_kernel(float* __restrict__ out, int n) {
    int i = blockIdx.x * blockDim.x + threadIdx.x;
    if (i < n) out[i] = 0.0f;
}

// counts[g] = (#nodes with seg==g); seg_ids are sorted ascending.
__global__ void counts_kernel(const int* __restrict__ seg, float* __restrict__ cnt,
                              int nodes, int G) {
    int g = blockIdx.x * blockDim.x + threadIdx.x;
    if (g >= G) return;
    // first index i with seg[i] > q
    auto ub = [&](int q) {
        int lo = 0, hi = nodes;
        while (lo < hi) {
            int mid = (lo + hi) >> 1;
            if (seg[mid] <= q) lo = mid + 1; else hi = mid;
        }
        return lo;
    };
    int hi = ub(g);
    int lo = (g == 0) ? 0 : ub(g - 1);
    cnt[g] = (float)(hi - lo);
}

__global__ void pool_main_kernel(const float* __restrict__ h,
                                 const int* __restrict__ seg,
                                 const float* __restrict__ W,
                                 const float* __restrict__ bias,
                                 const float* __restrict__ cnt,
                                 float* __restrict__ out,
                                 int nodes) {
    const int lane  = threadIdx.x & 31;
    const int m     = lane & 15;   // logical row within 16-node tile
    const int half  = lane >> 4;   // K-half for the A fragment
    const int waveId = (blockIdx.x * blockDim.x + threadIdx.x) >> 5;
    const int nWaves = (gridDim.x * blockDim.x) >> 5;

    // Effective weight vector Wt[32]: Wt[0]=0 (feature 0 is the value, not a
    // score feature), Wt[k]=W[k-1]. Fragment per lane: slice j covers
    // K = 4j + 2*half + {0,1}.
    float bw0[8], bw1[8];
#pragma unroll
    for (int j = 0; j < 8; ++j) {
        int idx = 4 * j + 2 * half;
        bw0[j] = (idx == 0) ? 0.0f : W[idx - 1];
        bw1[j] = W[idx];            // Wt[idx+1], idx+1 in [1,31]
    }
    const float bval = bias[0];

    for (long base = (long)waveId * 16; base < nodes; base += (long)nWaves * 16) {
        long row = base + m;
        if (row >= nodes) row = nodes - 1;          // tail clamp; masked later
        const float* p = h + row * 32;

        v8f c = {};
        float h0 = 0.0f;
#pragma unroll
        for (int j = 0; j < 8; ++j) {
#if __has_builtin(__builtin_nontemporal_load)
            v2f raw = __builtin_nontemporal_load((const v2f*)(p + 4 * j + 2 * half));
#else
            v2f raw = *(const v2f*)(p + 4 * j + 2 * half);
#endif
            if (j == 0 && half == 0) h0 = raw.x;    // raw value feature
            float e0 = raw.x > 0.0f ? raw.x : (__expf(raw.x) - 1.0f);  // ELU
            float e1 = raw.y > 0.0f ? raw.y : (__expf(raw.y) - 1.0f);
            v2f a;  a.x = e0 * bw0[j];  a.y = e1 * bw1[j];
            v2f bm; bm.x = 1.0f;        bm.y = 1.0f;   // B == 1 => D[m,n] = sum_k A[m,k]
            c = __builtin_amdgcn_wmma_f32_16x16x4_f32(
                    false, a, false, bm, (short)0, c, false, false);
        }

        // C layout: VGPR v, lanes 0-15 hold row v, lanes 16-31 hold row v+8,
        // and all columns are equal. Lane L wants score[L&15].
        float t = c[0];
        if ((lane & 7) == 1) t = c[1];
        if ((lane & 7) == 2) t = c[2];
        if ((lane & 7) == 3) t = c[3];
        if ((lane & 7) == 4) t = c[4];
        if ((lane & 7) == 5) t = c[5];
        if ((lane & 7) == 6) t = c[6];
        if ((lane & 7) == 7) t = c[7];
        int src = (lane < 8) ? lane : lane + 8;     // rows 8..15 live in lanes 16..23
        t = __shfl(t, src, 32);

        bool act = (half == 0) && (base + m < (long)nodes);
        int  s   = -1;
        float val = 0.0f;
        if (act) {
            s = seg[base + m];
            float n = cnt[s];
            if (n < 1.0f) n = 1.0f;
            float score = t + bval;
            float sig   = 1.0f / (1.0f + __expf(-score));
            float pm1   = 1.5f * sig - 0.5f;        // powers - 1
            val = h0 * EXP2F(__log2f(n) * pm1);
        }

        // Segmented inclusive scan over lanes 0..15 (segs sorted -> few runs),
        // then one atomic per run. Shuffles executed by all lanes (convergent).
        float acc = act ? val : 0.0f;
#pragma unroll
        for (int off = 1; off < 16; off <<= 1) {
            float o  = __shfl_up(acc, off, 32);
            int   so = __shfl_up(s,   off, 32);
            if (lane >= off && s >= 0 && so == s) acc += o;
        }
        int snext = __shfl_down(s, 1, 32);          // lane15 sees lane16's s==-1
        bool lastOfRun = act && (snext != s);
        if (lastOfRun) atomicAdd(&out[s], acc);
    }
}

extern "C" void kernel_launch(void* const* d_in, const int* in_sizes, int n_in,
                              void* d_out, int out_size, void* d_ws, size_t ws_size,
                              hipStream_t stream) {
    const float* h   = (const float*)d_in[0];
    const int*   seg = (const int*)d_in[1];
    const float* W   = (const float*)d_in[2];
    const float* b   = (const float*)d_in[3];
    (void)d_ws; (void)ws_size; (void)n_in;

    const int G     = out_size / 2;          // (output, counts) concatenated
    const int nodes = in_sizes[0] / 32;      // IN_FEATS = 32
    float* out = (float*)d_out;
    float* cnt = out + G;

    if (G <= 0 || nodes <= 0) return;

    zero_out_kernel<<<(G + 255) / 256, 256, 0, stream>>>(out, G);
    counts_kernel<<<(G + 255) / 256, 256, 0, stream>>>(seg, cnt, nodes, G);
    pool_main_kernel<<<4096, 256, 0, stream>>>(h, seg, W, b, cnt, out, nodes);
}